// BrainGNN_7069516169578
// MI455X (gfx1250) — compile-verified
//
#include <hip/hip_runtime.h>
#include <hip/hip_bf16.h>

typedef __attribute__((ext_vector_type(2))) float v2f;
typedef __attribute__((ext_vector_type(8))) float v8f;

#define NNODES 25600
#define RDIM   200
#define KDIM   8
#define D1DIM  32
#define NEDGES 819200
#define JDIM   1800        // R * (K+1)
#define NSTEP  450         // JDIM / 4
#define EPSF   1e-16f

// ---------------- Phase 0: h = relu(pseudo @ W1), [N,8] ----------------
__global__ __launch_bounds__(64) void hgen_kernel(const float* __restrict__ pseudo,
                                                  const float* __restrict__ W1,
                                                  float* __restrict__ h) {
    __shared__ float w1s[RDIM * KDIM];           // 6.4 KB
    const int t = threadIdx.x;
    for (int i = t; i < RDIM * KDIM; i += 64) w1s[i] = W1[i];
    __syncthreads();
    const int node = blockIdx.x * 8 + (t >> 3);
    const int k = t & 7;
    const float* p = pseudo + node * RDIM;
    float acc = 0.f;
    #pragma unroll 4
    for (int r = 0; r < RDIM; ++r) acc = fmaf(p[r], w1s[r * KDIM + k], acc);
    h[node * KDIM + k] = fmaxf(acc, 0.f);
}

// ---------------- Phase 0b: V[j,d] packing of W2/b2, [1800,32] ----------------
__global__ __launch_bounds__(256) void vgen_kernel(const float* __restrict__ W2,
                                                   const float* __restrict__ b2,
                                                   float* __restrict__ V) {
    const int t = blockIdx.x * 256 + threadIdx.x;
    if (t >= JDIM * D1DIM) return;
    const unsigned j = (unsigned)t >> 5, d = (unsigned)t & 31u;
    const unsigned r = j / 9u, k = j - r * 9u;
    V[t] = (k < 8u) ? W2[k * (RDIM * D1DIM) + r * D1DIM + d] : b2[r * D1DIM + d];
}

// ---------------- Phase 1: xt = Y @ V via V_WMMA_F32_16X16X4_F32 ----------------
// Y[n, r*9+k] = x[n,r]*hhat[n,k], hhat = [h0..h7, 1].  One wave per 16-node tile.
#define GEMM_WAVES 4
__global__ __launch_bounds__(32 * GEMM_WAVES) void gemm_xt_kernel(
        const float* __restrict__ x, const float* __restrict__ h,
        const float* __restrict__ V, float* __restrict__ xt) {
    __shared__ float xs[GEMM_WAVES][16][RDIM];   // 51.2 KB of the 320 KB WGP LDS
    __shared__ float hs[GEMM_WAVES][16][9];
    const int lane = threadIdx.x & 31;
    const int w    = threadIdx.x >> 5;
    const int base = (blockIdx.x * GEMM_WAVES + w) * 16;

    // stage x tile (coalesced by row) and augmented h row
    for (int m = 0; m < 16; ++m)
        for (int r = lane; r < RDIM; r += 32)
            xs[w][m][r] = x[(base + m) * RDIM + r];
    if (lane < 16) {
        #pragma unroll
        for (int k = 0; k < 8; ++k) hs[w][lane][k] = h[(base + lane) * KDIM + k];
        hs[w][lane][8] = 1.0f;
    }
    __syncthreads();

    const int m    = lane & 15;       // A: row M; B: col N
    const int half = lane >> 4;       // lanes 16-31 hold K offsets +2,+3
    v8f c0 = {}; v8f c1 = {};
    for (int step = 0; step < NSTEP; ++step) {
        const unsigned j0 = (unsigned)(step * 4 + half * 2);
        const unsigned j1 = j0 + 1u;
        const unsigned r0 = j0 / 9u, k0 = j0 - r0 * 9u;
        const unsigned r1 = j1 / 9u, k1 = j1 - r1 * 9u;
        v2f a, b0, b1;
        a.x  = xs[w][m][r0] * hs[w][m][k0];
        a.y  = xs[w][m][r1] * hs[w][m][k1];
        b0.x = V[j0 * D1DIM + m];
        b0.y = V[j1 * D1DIM + m];
        b1.x = V[j0 * D1DIM + 16 + m];
        b1.y = V[j1 * D1DIM + 16 + m];
        c0 = __builtin_amdgcn_wmma_f32_16x16x4_f32(false, a, false, b0, (short)0, c0, false, false);
        c1 = __builtin_amdgcn_wmma_f32_16x16x4_f32(false, a, false, b1, (short)0, c1, false, false);
    }
    // D layout: VGPR i -> M=i (lanes 0-15) / M=8+i (lanes 16-31), N = lane&15
    #pragma unroll
    for (int i = 0; i < 8; ++i) {
        const int node = base + half * 8 + i;
        xt[node * D1DIM + m]      = c0[i];
        xt[node * D1DIM + 16 + m] = c1[i];
    }
}

// ---------------- Phase 2: segment softmax stats over dst ----------------
__global__ __launch_bounds__(256) void seg_init_kernel(unsigned* __restrict__ maxw_u,
                                                       float* __restrict__ sum) {
    const int n = blockIdx.x * 256 + threadIdx.x;
    if (n < NNODES) { maxw_u[n] = __float_as_uint(1.0f); sum[n] = 0.f; }  // self loop ew=1
}

__global__ __launch_bounds__(256) void seg_max_kernel(const int* __restrict__ dst,
                                                      const float* __restrict__ ew,
                                                      unsigned* __restrict__ maxw_u) {
    const int e = blockIdx.x * 256 + threadIdx.x;
    if (e < NEDGES) atomicMax(&maxw_u[dst[e]], __float_as_uint(ew[e]));  // ew >= 0
}

__global__ __launch_bounds__(256) void seg_sum_kernel(const int* __restrict__ dst,
                                                      const float* __restrict__ ew,
                                                      const unsigned* __restrict__ maxw_u,
                                                      float* __restrict__ sum) {
    const int t = blockIdx.x * 256 + threadIdx.x;
    if (t < NEDGES) {
        const int d = dst[t];
        atomicAdd(&sum[d], expf(ew[t] - __uint_as_float(maxw_u[d])));
    } else if (t < NEDGES + NNODES) {
        const int n = t - NEDGES;          // self loop contribution
        atomicAdd(&sum[n], expf(1.0f - __uint_as_float(maxw_u[n])));
    }
}

// ---------------- Phase 3: out = bias + alpha_self * xt (pure write) ----------------
__global__ __launch_bounds__(256) void out_init_kernel(const float* __restrict__ xt,
                                                       const unsigned* __restrict__ maxw_u,
                                                       const float* __restrict__ sum,
                                                       const float* __restrict__ bias,
                                                       float* __restrict__ out) {
    const int t = blockIdx.x * 256 + threadIdx.x;
    if (t >= NNODES * D1DIM) return;
    const int n = t >> 5, d = t & 31;
    const float a = expf(1.0f - __uint_as_float(maxw_u[n])) / (sum[n] + EPSF);
    out[t] = bias[d] + a * xt[t];
}

// ---------------- Phase 4: scatter alpha * xt[src] into out[dst] ----------------
__global__ __launch_bounds__(256) void scatter_kernel(const int* __restrict__ src,
                                                      const int* __restrict__ dst,
                                                      const float* __restrict__ ew,
                                                      const unsigned* __restrict__ maxw_u,
                                                      const float* __restrict__ sum,
                                                      const float* __restrict__ xt,
                                                      float* __restrict__ out) {
    const int t = blockIdx.x * 256 + threadIdx.x;    // one wave32 per edge
    const int e = t >> 5, d = t & 31;
    if (e >= NEDGES) return;
    const int dn = dst[e], sn = src[e];
    const float alpha = expf(ew[e] - __uint_as_float(maxw_u[dn])) / (sum[dn] + EPSF);
    atomicAdd(&out[dn * D1DIM + d], alpha * xt[sn * D1DIM + d]);
}

extern "C" void kernel_launch(void* const* d_in, const int* in_sizes, int n_in,
                              void* d_out, int out_size, void* d_ws, size_t ws_size,
                              hipStream_t stream) {
    const float* x      = (const float*)d_in[0];
    const float* pseudo = (const float*)d_in[1];
    const int*   ei     = (const int*)d_in[2];
    const float* ew     = (const float*)d_in[3];
    const float* W1     = (const float*)d_in[4];
    const float* W2     = (const float*)d_in[5];
    const float* b2     = (const float*)d_in[6];
    const float* bias   = (const float*)d_in[7];
    float* out          = (float*)d_out;
    const int* src = ei;
    const int* dst = ei + NEDGES;

    // workspace layout (bytes, 256-aligned)
    char* ws = (char*)d_ws;
    float*    h      = (float*)(ws + 0);           // N*8   = 819200 B
    float*    V      = (float*)(ws + 819200);      // 1800*32*4 = 230400 B
    float*    xt     = (float*)(ws + 1049600);     // N*32*4 = 3276800 B
    unsigned* maxw_u = (unsigned*)(ws + 4326400);  // N*4
    float*    sum    = (float*)(ws + 4428800);     // N*4   (end 4531200 B)

    hgen_kernel<<<NNODES / 8, 64, 0, stream>>>(pseudo, W1, h);
    vgen_kernel<<<(JDIM * D1DIM + 255) / 256, 256, 0, stream>>>(W2, b2, V);
    seg_init_kernel<<<(NNODES + 255) / 256, 256, 0, stream>>>(maxw_u, sum);

    gemm_xt_kernel<<<NNODES / (16 * GEMM_WAVES), 32 * GEMM_WAVES, 0, stream>>>(x, h, V, xt);

    seg_max_kernel<<<(NEDGES + 255) / 256, 256, 0, stream>>>(dst, ew, maxw_u);
    seg_sum_kernel<<<(NEDGES + NNODES + 255) / 256, 256, 0, stream>>>(dst, ew, maxw_u, sum);

    out_init_kernel<<<(NNODES * D1DIM + 255) / 256, 256, 0, stream>>>(xt, maxw_u, sum, bias, out);
    scatter_kernel<<<(NEDGES * 32 + 255) / 256, 256, 0, stream>>>(src, dst, ew, maxw_u, sum, xt, out);
}